// ECMoEGate_43121471652482
// MI455X (gfx1250) — compile-verified
//
#include <hip/hip_runtime.h>
#include <hip/hip_bf16.h>

typedef __attribute__((ext_vector_type(2))) float v2f;
typedef __attribute__((ext_vector_type(8))) float v8f;

#define HDIM 2048
#define NTOK 16384   // 4 * 4096
#define NEXP 16
#define CAP  2048    // ceil(NTOK/NEXP * 2)

// ---------------------------------------------------------------------------
// Kernel 1: logits = x @ W^T via V_WMMA_F32_16X16X4_F32, fused sigmoid,
// stored expert-major (scoresT[E][NTOK]) for the selection pass.
// One wave handles a 16-token x 16-expert tile; 8 waves per block.
// ---------------------------------------------------------------------------
__global__ __launch_bounds__(256) void gate_gemm_sigmoid(
    const float* __restrict__ x,      // [NTOK, HDIM]
    const float* __restrict__ w,      // [NEXP, HDIM]
    float* __restrict__ scoresT)      // [NEXP, NTOK]
{
  const int lane = threadIdx.x & 31;
  const int wave = threadIdx.x >> 5;
  const int tile = blockIdx.x * 8 + wave;   // 16-token tile
  const int tok0 = tile * 16;

  // f32 16x16x4 A fragment: lanes 0-15 -> M=lane, K={0,1}; lanes 16-31 -> M=lane-16, K={2,3}
  const int row   = lane & 15;
  const int khalf = (lane >> 4) << 1;       // 0 or 2
  const float* __restrict__ xrow = x + (size_t)(tok0 + row) * HDIM + khalf;
  // B fragment (4x16): column N = lane&15 = expert row of W, same K split
  const float* __restrict__ wrow = w + (size_t)row * HDIM + khalf;

  v8f c0 = {};
  v8f c1 = {};
  for (int k0 = 0; k0 < HDIM; k0 += 32) {
    __builtin_prefetch(xrow + k0 + 512, 0, 0);   // global_prefetch_b8, ~2KB ahead
#pragma unroll
    for (int kk = 0; kk < 32; kk += 8) {
      v2f a0 = *(const v2f*)(xrow + k0 + kk);
      v2f b0 = *(const v2f*)(wrow + k0 + kk);
      c0 = __builtin_amdgcn_wmma_f32_16x16x4_f32(false, a0, false, b0,
                                                 (short)0, c0, false, false);
      v2f a1 = *(const v2f*)(xrow + k0 + kk + 4);
      v2f b1 = *(const v2f*)(wrow + k0 + kk + 4);
      c1 = __builtin_amdgcn_wmma_f32_16x16x4_f32(false, a1, false, b1,
                                                 (short)0, c1, false, false);
    }
  }
  v8f c = c0 + c1;

  // C/D layout: VGPR r, lanes 0-15 -> M=r, lanes 16-31 -> M=8+r; N = lane&15
  const int e     = lane & 15;
  const int mbase = (lane >> 4) * 8;
#pragma unroll
  for (int r = 0; r < 8; ++r) {
    float s = 1.0f / (1.0f + __expf(-c[r]));
    scoresT[(size_t)e * NTOK + (tok0 + mbase + r)] = s;
  }
}

// ---------------------------------------------------------------------------
// Kernel 2: per-expert top-2048 of 16384, sorted descending.
// Keys: (score_bits << 32) | ~id  (sigmoid > 0 => uint-monotone; ~id makes
// ties prefer the smaller token index). Running top-2048 list in LDS, merged
// with the 7 other sorted chunks via the bitonic top-k max-combine.
// ---------------------------------------------------------------------------
__device__ __forceinline__ unsigned long long pack_key(float v, int id) {
  return ((unsigned long long)__float_as_uint(v) << 32) |
         (unsigned long long)(unsigned int)(~id);
}

__device__ __forceinline__ void bitonic_sort_desc(unsigned long long* a) {
  const int tid = threadIdx.x;   // 1024 threads, 2048 elements
  for (int k = 2; k <= CAP; k <<= 1) {
    for (int j = k >> 1; j > 0; j >>= 1) {
      __syncthreads();
#pragma unroll 1
      for (int i = tid; i < CAP; i += 1024) {
        int l = i ^ j;
        if (l > i) {
          unsigned long long ai = a[i], al = a[l];
          bool asc  = (i & k) != 0;               // descending overall
          bool swp  = asc ? (ai > al) : (ai < al);
          if (swp) { a[i] = al; a[l] = ai; }
        }
      }
    }
  }
  __syncthreads();
}

__global__ __launch_bounds__(1024) void expert_topk(
    const float* __restrict__ scoresT,  // [NEXP, NTOK]
    int* __restrict__ ids_out,          // [NEXP, CAP]
    float* __restrict__ sc_out)         // [NEXP, CAP]
{
  __shared__ unsigned long long run[CAP];
  __shared__ unsigned long long buf[CAP];
  const int e   = blockIdx.x;
  const int tid = threadIdx.x;
  const float* __restrict__ s = scoresT + (size_t)e * NTOK;

  // chunk 0 -> running list
#pragma unroll 1
  for (int i = tid; i < CAP; i += 1024) run[i] = pack_key(s[i], i);
  bitonic_sort_desc(run);

  for (int c = 1; c < NTOK / CAP; ++c) {
    const int base = c * CAP;
#pragma unroll 1
    for (int i = tid; i < CAP; i += 1024) buf[i] = pack_key(s[base + i], base + i);
    bitonic_sort_desc(buf);

    // top-2048 of (run ∪ buf): elementwise max against reversed buf -> bitonic
#pragma unroll 1
    for (int i = tid; i < CAP; i += 1024) {
      unsigned long long a0 = run[i];
      unsigned long long b0 = buf[(CAP - 1) - i];
      run[i] = a0 > b0 ? a0 : b0;
    }
    // clean descending bitonic merge
    for (int j = CAP / 2; j > 0; j >>= 1) {
      __syncthreads();
#pragma unroll 1
      for (int i = tid; i < CAP; i += 1024) {
        int l = i ^ j;
        if (l > i) {
          unsigned long long ai = run[i], al = run[l];
          if (ai < al) { run[i] = al; run[l] = ai; }
        }
      }
    }
    __syncthreads();
  }

#pragma unroll 1
  for (int i = tid; i < CAP; i += 1024) {
    unsigned long long v = run[i];
    ids_out[e * CAP + i] = (int)(~(unsigned int)v);
    sc_out[e * CAP + i]  = __uint_as_float((unsigned int)(v >> 32));
  }
}

// ---------------------------------------------------------------------------
extern "C" void kernel_launch(void* const* d_in, const int* in_sizes, int n_in,
                              void* d_out, int out_size, void* d_ws, size_t ws_size,
                              hipStream_t stream) {
  const float* x = (const float*)d_in[0];   // hidden_states [4,4096,2048] f32
  const float* w = (const float*)d_in[1];   // weight [16,2048] f32

  float* scoresT = (float*)d_ws;            // [16, 16384] f32 = 1 MB scratch

  int*   ids_out = (int*)d_out;                      // [16,2048] int32 (bitwise)
  float* sc_out  = (float*)d_out + NEXP * CAP;       // [16,2048] f32

  // 1024 tiles of 16 tokens, 8 waves (tiles) per 256-thread block
  gate_gemm_sigmoid<<<NTOK / 16 / 8, 256, 0, stream>>>(x, w, scoresT);
  expert_topk<<<NEXP, 1024, 0, stream>>>(scoresT, ids_out, sc_out);
}